// VectorQuantizer_19121194402367
// MI455X (gfx1250) — compile-verified
//
#include <hip/hip_runtime.h>

typedef __attribute__((ext_vector_type(2))) float v2f;
typedef __attribute__((ext_vector_type(4))) float v4f;
typedef __attribute__((ext_vector_type(8))) float v8f;

constexpr int Bn = 16384, Gn = 8, Kn = 512, Dn = 256;
constexpr int BM = 32;     // batch rows per block
constexpr int KC = 64;     // codebook columns per chunk
constexpr int DC = 128;    // depth per LDS stage
constexpr int LSTR = 132;  // padded LDS row stride (floats); 132 % 64 banks = 4 -> conflict-free b64 frag loads

// Async global->LDS copy of 16 bytes per lane (CDNA5, tracked by ASYNCcnt).
// ldsoff = per-lane LDS byte offset (low 32 bits of the generic shared pointer,
// per the flat aperture rule).
__device__ __forceinline__ void async_copy_b128(unsigned ldsoff, const float* src) {
  asm volatile("global_load_async_to_lds_b128 %0, %1, off"
               :: "v"(ldsoff), "v"(src) : "memory");
}
__device__ __forceinline__ void wait_async0() {
  asm volatile("s_wait_asynccnt 0x0" ::: "memory");
}

// ---------------- kernel 1: e2 = ||e||^2 per code row, zero accumulators ----------------
__global__ __launch_bounds__(256) void vq_prep(const float* __restrict__ emb,
                                               float* __restrict__ e2,
                                               float* __restrict__ sse,
                                               unsigned* __restrict__ counts) {
  const int lane = threadIdx.x & 31;
  const int row  = blockIdx.x * 8 + (threadIdx.x >> 5);   // 4096 rows over 512 blocks
  if (row < Gn * Kn) {
    const float* p = emb + (size_t)row * Dn;
    float s = 0.f;
    for (int d = lane; d < Dn; d += 32) { float v = p[d]; s += v * v; }
    #pragma unroll
    for (int off = 16; off; off >>= 1) s += __shfl_xor(s, off, 32);
    if (lane == 0) e2[row] = s;
  }
  const int t = blockIdx.x * 256 + threadIdx.x;
  if (t < Gn * Kn) counts[t] = 0u;
  if (t < Gn) sse[t] = 0.f;
}

// ---------------- kernel 2: batched GEMM (WMMA f32) + argmin + gather + outputs ----------------
__global__ __launch_bounds__(256) void vq_main(const float* __restrict__ x,      // [B,G,D]
                                               const float* __restrict__ emb,    // [G,K,D]
                                               const float* __restrict__ e2,     // [G,K]
                                               float* __restrict__ q_all,        // [B,G,D]
                                               float* __restrict__ q_gbd,        // [G,B,D]
                                               float* __restrict__ enc,          // [G,B,K]
                                               float* __restrict__ sse,          // [G]
                                               unsigned* __restrict__ counts) {  // [G,K]
  __shared__ float xs[BM * LSTR];
  __shared__ float es[KC * LSTR];
  __shared__ float redv[4][BM];
  __shared__ int   redi[4][BM];
  __shared__ int   fidx[BM];
  __shared__ float rowsum[BM];

  const int g    = blockIdx.y;
  const int b0   = blockIdx.x * BM;
  const int tid  = threadIdx.x;
  const int lane = tid & 31;
  const int wave = tid >> 5;
  const int rowg = wave & 1;     // 2 row-groups of 16
  const int colg = wave >> 1;    // 4 col-groups of 16 within a 64-col chunk
  const int m0   = rowg * 16;
  const int n0   = colg * 16;
  const int l15  = lane & 15;
  const int hi   = lane >> 4;    // 0/1: K-pair selector of the 16x4 A frag / 4x16 B frag

  const float* xg  = x   + ((size_t)b0 * Gn + g) * Dn;   // row stride G*D
  const float* eg  = emb + (size_t)g * Kn * Dn;
  const float* e2g = e2  + (size_t)g * Kn;

  const unsigned xs_base = (unsigned)(uintptr_t)&xs[0];
  const unsigned es_base = (unsigned)(uintptr_t)&es[0];

  float minv[8];
  int   mini[8];
  #pragma unroll
  for (int r = 0; r < 8; ++r) { minv[r] = 3.4e38f; mini[r] = 0; }

  for (int kc = 0; kc < Kn / KC; ++kc) {
    v8f acc = {0.f, 0.f, 0.f, 0.f, 0.f, 0.f, 0.f, 0.f};
    for (int dc = 0; dc < Dn / DC; ++dc) {
      __syncthreads();                 // previous stage fully consumed
      // stage x tile: BM x DC = 1024 float4 -> 4 async b128 per thread, all in flight
      #pragma unroll
      for (int it = 0; it < (BM * DC) / 4 / 256; ++it) {
        int i = tid + it * 256;
        int r = i >> 5;                // DC/4 = 32 float4 per row
        int c = (i & 31) << 2;
        async_copy_b128(xs_base + (unsigned)((r * LSTR + c) * 4),
                        xg + (size_t)r * (Gn * Dn) + dc * DC + c);
      }
      // stage e tile: KC x DC = 2048 float4 -> 8 async b128 per thread
      const float* ebase = eg + (size_t)(kc * KC) * Dn + dc * DC;
      #pragma unroll
      for (int it = 0; it < (KC * DC) / 4 / 256; ++it) {
        int i = tid + it * 256;
        int r = i >> 5;
        int c = (i & 31) << 2;
        async_copy_b128(es_base + (unsigned)((r * LSTR + c) * 4),
                        ebase + (size_t)r * Dn + c);
      }
      wait_async0();                   // this wave's copies landed in LDS
      __syncthreads();                 // whole tile visible to all waves
      // 16x16 tile accumulation over this D stage: K-step of 4
      #pragma unroll 8
      for (int kk = 0; kk < DC; kk += 4) {
        // A 16x4: lanes 0-15 rows M, K={kk,kk+1}; lanes 16-31 same rows, K={kk+2,kk+3}
        v2f a = *(const v2f*)&xs[(m0 + l15) * LSTR + kk + (hi << 1)];
        // B 4x16 = E^T: column n holds codebook row n
        v2f b = *(const v2f*)&es[(n0 + l15) * LSTR + kk + (hi << 1)];
        acc = __builtin_amdgcn_wmma_f32_16x16x4_f32(false, a, false, b,
                                                    (short)0, acc, false, false);
      }
    }
    // scores: e2[k] - 2*xe ; running argmin (k increases monotonically -> strict < keeps first)
    const int   kcol = kc * KC + n0 + l15;
    const float e2v  = e2g[kcol];
    #pragma unroll
    for (int r = 0; r < 8; ++r) {
      float s = __builtin_fmaf(-2.f, acc[r], e2v);
      if (s < minv[r]) { minv[r] = s; mini[r] = kcol; }
    }
  }

  // reduce argmin across the 16 lanes of each half (C layout: VGPR r -> rows m0+r / m0+8+r)
  #pragma unroll
  for (int r = 0; r < 8; ++r) {
    float v = minv[r]; int i = mini[r];
    #pragma unroll
    for (int off = 8; off; off >>= 1) {
      float ov = __shfl_xor(v, off, 32);
      int   oi = __shfl_xor(i, off, 32);
      if (ov < v || (ov == v && oi < i)) { v = ov; i = oi; }
    }
    if (l15 == 0) {
      int row = m0 + r + (hi << 3);
      redv[colg][row] = v;
      redi[colg][row] = i;
    }
  }
  __syncthreads();
  if (tid < BM) {
    float v = redv[0][tid]; int i = redi[0][tid];
    #pragma unroll
    for (int c = 1; c < 4; ++c) {
      float ov = redv[c][tid]; int oi = redi[c][tid];
      if (ov < v || (ov == v && oi < i)) { v = ov; i = oi; }
    }
    fidx[tid] = i;
  }
  __syncthreads();

  // epilogue: 8 threads per row -> gather code row, both quantized layouts, one-hot, SSE, counts
  {
    const int row = tid >> 3;
    const int sub = tid & 7;
    const int b   = b0 + row;
    const int idx = fidx[row];
    const float* qrow = eg + (size_t)idx * Dn;
    const float* xrow = x + ((size_t)b * Gn + g) * Dn;
    float* qa = q_all + ((size_t)b * Gn + g) * Dn;
    float* qg = q_gbd + ((size_t)g * Bn + b) * Dn;
    float ds = 0.f;
    #pragma unroll
    for (int j = 0; j < 8; ++j) {          // 8 float4 = 32 floats per thread
      int d = (sub * 8 + j) * 4;
      v4f qv = *(const v4f*)(qrow + d);
      v4f xv = *(const v4f*)(xrow + d);
      *(v4f*)(qa + d) = qv;
      *(v4f*)(qg + d) = qv;
      v4f df = qv - xv;
      ds += df.x * df.x + df.y * df.y + df.z * df.z + df.w * df.w;
    }
    #pragma unroll
    for (int off = 1; off < 8; off <<= 1) ds += __shfl_xor(ds, off, 32);
    if (sub == 0) {
      rowsum[row] = ds;
      atomicAdd(&counts[(size_t)g * Kn + idx], 1u);
    }
    // one-hot row: 512 floats / 8 threads = 16 float4 each
    float* erow = enc + ((size_t)g * Bn + b) * Kn + sub * 64;
    const int rel = idx - sub * 64;
    #pragma unroll
    for (int j = 0; j < 16; ++j) {
      v4f z = {0.f, 0.f, 0.f, 0.f};
      int base = j * 4;
      if (rel >= base && rel < base + 4) ((float*)&z)[rel - base] = 1.0f;
      *(v4f*)(erow + base) = z;
    }
  }
  __syncthreads();
  if (tid == 0) {
    float s = 0.f;
    for (int r = 0; r < BM; ++r) s += rowsum[r];
    atomicAdd(&sse[g], s);      // one atomic per block
  }
}

// ---------------- kernel 3: loss + perplexity scalars ----------------
__global__ __launch_bounds__(256) void vq_final(const float* __restrict__ sse,
                                                const unsigned* __restrict__ counts,
                                                float* __restrict__ out_loss,
                                                float* __restrict__ out_perp) {
  __shared__ float red[256];
  const int tid = threadIdx.x;
  float perp = 0.f;
  for (int g = 0; g < Gn; ++g) {
    float s = 0.f;
    for (int k = tid; k < Kn; k += 256) {
      float p = (float)counts[g * Kn + k] * (1.0f / (float)Bn);
      s += p * logf(p + 1e-10f);
    }
    red[tid] = s;
    __syncthreads();
    for (int off = 128; off; off >>= 1) {
      if (tid < off) red[tid] += red[tid + off];
      __syncthreads();
    }
    if (tid == 0) perp += expf(-red[0]);
    __syncthreads();
  }
  if (tid == 0) {
    *out_perp = perp / (float)Gn;
    float loss = 0.f;
    for (int g = 0; g < Gn; ++g)
      loss += 1.25f * sse[g] / ((float)Bn * (float)Dn);   // q_latent + 0.25*e_latent, both == MSE
    *out_loss = loss / (float)Gn;
  }
}

extern "C" void kernel_launch(void* const* d_in, const int* in_sizes, int n_in,
                              void* d_out, int out_size, void* d_ws, size_t ws_size,
                              hipStream_t stream) {
  (void)in_sizes; (void)n_in; (void)out_size; (void)ws_size;
  const float* x   = (const float*)d_in[0];   // [B,G,D]
  const float* emb = (const float*)d_in[1];   // [G,K,D]

  float* out    = (float*)d_out;
  float* o_loss = out;                                    // [1]
  float* o_qall = out + 1;                                // [B,G,D]
  float* o_perp = o_qall + (size_t)Bn * Gn * Dn;          // [1]
  float* o_qgbd = o_perp + 1;                             // [G,B,D]
  float* o_enc  = o_qgbd + (size_t)Gn * Bn * Dn;          // [G,B,K]

  float*    e2     = (float*)d_ws;                        // [G*K]
  float*    sse    = e2 + Gn * Kn;                        // [G]
  unsigned* counts = (unsigned*)(sse + Gn);               // [G*K]

  vq_prep<<<512, 256, 0, stream>>>(emb, e2, sse, counts);
  dim3 grid(Bn / BM, Gn);
  vq_main<<<grid, 256, 0, stream>>>(x, emb, e2, o_qall, o_qgbd, o_enc, sse, counts);
  vq_final<<<1, 256, 0, stream>>>(sse, counts, o_loss, o_perp);
}